// GaussianBlur2D_34986803593577
// MI455X (gfx1250) — compile-verified
//
#include <hip/hip_runtime.h>
#include <hip/hip_bf16.h>
#include <math.h>

// GaussianBlur2D on MI455X (gfx1250, wave32):
//  - separable 11x11 Gaussian, fused H+V passes through LDS
//  - each 16x16 tile of each 1D pass = banded 16x28 @ 28x16 matmul
//    -> 7x V_WMMA_F32_16X16X4_F32 per tile per pass (63 per wave)
//  - memory-bound design: 1 HBM read + 1 HBM write per element (~92us floor)
//  - even LDS strides so every WMMA B-operand chunk is one ds_load_b64

typedef __attribute__((ext_vector_type(2))) float v2f;
typedef __attribute__((ext_vector_type(8))) float v8f;

#define KS    11
#define RAD   5
#define HH    512
#define WW    512
#define TILE  64              // 64x64 output tile per workgroup
#define HALO  (TILE + 2*RAD)  // 74
#define XSTR  76              // even stride: 8B-aligned pairs, conflict-free (12r mod 64)
#define MSTR  76              // transposed intermediate: [64 cols][74 rows]

__global__ __launch_bounds__(128)
void gauss_blur_sep_wmma(const float* __restrict__ x,
                         const float* __restrict__ sigma,
                         float* __restrict__ out)
{
    __shared__ float xs[HALO][XSTR];   // input tile + halo         (22496 B)
    __shared__ float mt[TILE][MSTR];   // H-pass result, [col][row] (19456 B)
    __shared__ float gk[16];           // 1D gaussian coeffs (11 used)

    const int t    = threadIdx.x;      // 0..127
    const int lane = t & 31;
    const int wave = t >> 5;           // 0..3 : one 16-col strip per wave

    // ---- block -> (plane, tile) mapping: 8x8 tiles of 64x64 per 512x512 plane
    const int plane = blockIdx.x >> 6;
    const int tile  = blockIdx.x & 63;
    const int tr = (tile >> 3) * TILE;
    const int tc = (tile & 7) * TILE;
    const float* __restrict__ src = x   + (size_t)plane * (HH * WW);
    float* __restrict__       dst = out + (size_t)plane * (HH * WW);

    // ---- 1D gaussian from sigma:  g = e / sum(e),  e_i = exp(-(i-5)^2/(2 s^2))
    if (t < 16) {
        float s = fabsf(sigma[0]) + 1e-6f;
        float inv2s2 = 1.0f / (2.0f * s * s);
        float sum = 0.0f, mine = 0.0f;
        #pragma unroll
        for (int i = 0; i < KS; ++i) {
            float r = (float)i - 5.0f;
            float e = __expf(-r * r * inv2s2);
            sum += e;
            if (i == t) mine = e;
        }
        gk[t] = (t < KS) ? (mine / sum) : 0.0f;
    }

    // ---- load 74x74 halo tile with reflect indexing (coalesced row-major)
    for (int i = t; i < HALO * HALO; i += 128) {
        int rr = i / HALO, cc = i - rr * HALO;
        int gr = tr + rr - RAD;
        int gc = tc + cc - RAD;
        gr = (gr < 0) ? -gr : ((gr >= HH) ? (2 * HH - 2 - gr) : gr);
        gc = (gc < 0) ? -gc : ((gc >= WW) ? (2 * WW - 2 - gc) : gc);
        xs[rr][cc] = src[gr * WW + gc];
    }
    __syncthreads();

    // ---- constant band matrix A (16x28, K-chunks of 4), shared by both passes
    // ISA A layout (16x4 f32): lane M = lane&15; vgpr0 = K {0 | 2}, vgpr1 = K {1 | 3}
    const int nlow = lane & 15;
    const int kb   = (lane >> 4) << 1;      // 0 for lanes 0-15, 2 for 16-31
    const int mofs = (lane >> 4) << 3;      // D-matrix M offset: 0 or 8
    v2f aband[7];
    #pragma unroll
    for (int j = 0; j < 7; ++j) {
        int k0 = 4 * j + kb;
        int d0 = k0 - nlow, d1 = k0 + 1 - nlow;
        aband[j].x = (d0 >= 0 && d0 <= 10) ? gk[d0] : 0.0f;
        aband[j].y = (d1 >= 0 && d1 <= 10) ? gk[d1] : 0.0f;
    }

    const int s0 = wave * 16;   // this wave's 16-wide column strip

    // ---- horizontal pass (transposed output):  OutT[c,r] = sum_m A[c,m] * xs[r][s0+m]
    // rows 0..73 of xs needed -> 5 row-tiles of 16 (rows >= 74 discarded)
    for (int rt = 0; rt < 5; ++rt) {
        int rowbase = rt * 16;
        int brow = rowbase + nlow; if (brow > 73) brow = 73;   // clamp (discarded later)
        v8f acc = {};
        #pragma unroll
        for (int j = 0; j < 7; ++j) {
            int c0 = s0 + 4 * j + kb;        // even by construction
            if (c0 > 73) c0 = 72;            // K=26,27 have zero coeff; keep even+valid
            v2f b = *(const v2f*)&xs[brow][c0];   // one ds_load_b64
            acc = __builtin_amdgcn_wmma_f32_16x16x4_f32(
                      false, aband[j], false, b, (short)0, acc, false, false);
        }
        // D: M = column-in-strip (v + mofs), N = row (nlow); store transposed
        int orow = rowbase + nlow;
        if (orow <= 73) {
            #pragma unroll
            for (int v = 0; v < 8; ++v)
                mt[s0 + v + mofs][orow] = acc[v];
        }
    }
    __syncthreads();

    // ---- vertical pass: out[r,c] = sum_m A[r,m] * mt[c][rowbase+m]
    for (int rt = 0; rt < 4; ++rt) {
        int rowbase = rt * 16;
        v8f acc = {};
        #pragma unroll
        for (int j = 0; j < 7; ++j) {
            int p0 = rowbase + 4 * j + kb;   // even by construction
            if (p0 > 73) p0 = 72;            // zero-coeff K pad; keep even+valid
            v2f b = *(const v2f*)&mt[s0 + nlow][p0];   // one ds_load_b64
            acc = __builtin_amdgcn_wmma_f32_16x16x4_f32(
                      false, aband[j], false, b, (short)0, acc, false, false);
        }
        // D: M = out row (v + mofs), N = out col (nlow)
        #pragma unroll
        for (int v = 0; v < 8; ++v) {
            int orow = tr + rowbase + v + mofs;
            int ocol = tc + s0 + nlow;
            dst[(size_t)orow * WW + ocol] = acc[v];
        }
    }
}

extern "C" void kernel_launch(void* const* d_in, const int* in_sizes, int n_in,
                              void* d_out, int out_size, void* d_ws, size_t ws_size,
                              hipStream_t stream) {
    const float* x     = (const float*)d_in[0];
    const float* sigma = (const float*)d_in[1];
    float* out         = (float*)d_out;

    const int planes = in_sizes[0] / (HH * WW);            // B*C = 1024
    const int tiles_per_plane = (HH / TILE) * (WW / TILE); // 64
    dim3 grid(planes * tiles_per_plane);
    dim3 block(128);
    gauss_blur_sep_wmma<<<grid, block, 0, stream>>>(x, sigma, out);
}